// GIN_70282844831797
// MI455X (gfx1250) — compile-verified
//
#include <hip/hip_runtime.h>
#include <hip/hip_bf16.h>

#define N_NODES 100000
#define N_EDGES 1600000
#define DIM 128
#define BN_EPS 1e-5f
#define N_TILES (N_NODES / 16)   // 6250, exact

typedef __attribute__((ext_vector_type(2))) float v2f;
typedef __attribute__((ext_vector_type(8))) float v8f;

// Cap the backend scheduling window so each K-step stays a tight
// {4 ds_load_b64 -> 4 v_wmma} group.
#if defined(__has_builtin)
#if __has_builtin(__builtin_amdgcn_sched_barrier)
#define SCHED_FENCE() __builtin_amdgcn_sched_barrier(0)
#endif
#endif
#ifndef SCHED_FENCE
#define SCHED_FENCE()
#endif

// ---------------------------------------------------------------------------
// Zero a float region before atomic accumulation.
// ---------------------------------------------------------------------------
__global__ void gin_zero_kernel(float* __restrict__ p, long long n) {
    long long i = (long long)blockIdx.x * blockDim.x + threadIdx.x;
    if (i < n) p[i] = 0.0f;
}

// ---------------------------------------------------------------------------
// Edge scatter: s[dst] += x[src] (per feature), cnt[dst] += 1 (layer 0 only;
// degrees are identical across layers). One thread per (edge, 4 features):
// one global_load_b128 gather feeding 4 f32 atomics. Consecutive 32 threads
// cover one edge's full 128-float row -> fully coalesced gather.
// ---------------------------------------------------------------------------
__global__ void gin_scatter_kernel(const float* __restrict__ xin,
                                   const int* __restrict__ src,
                                   const int* __restrict__ dst,
                                   float* __restrict__ s,
                                   float* __restrict__ cnt,
                                   int do_cnt) {
    long long i = (long long)blockIdx.x * blockDim.x + threadIdx.x;
    if (i >= (long long)N_EDGES * (DIM / 4)) return;
    int e  = (int)(i >> 5);          // edge index
    int c4 = (int)(i & 31);          // which float4 of the row
    int se = src[e];
    int de = dst[e];
    float4 xv = ((const float4*)(xin + (long long)se * DIM))[c4];
    float* srow = s + (long long)de * DIM + c4 * 4;
    atomicAdd(srow + 0, xv.x);
    atomicAdd(srow + 1, xv.y);
    atomicAdd(srow + 2, xv.z);
    atomicAdd(srow + 3, xv.w);
    if (do_cnt && c4 == 0) atomicAdd(&cnt[de], 1.0f);
}

// ---------------------------------------------------------------------------
// Stage W into LDS K-pair-packed: shWp[j*DIM + col] = (W[2j][col], W[2j+1][col]).
// Each WMMA B fragment (K rows k0,k0+1 at one column) becomes a single
// aligned ds_load_b64 straight into the operand register pair.
// ---------------------------------------------------------------------------
__device__ __forceinline__ void stage_w_packed(v2f* shWp, const float* __restrict__ W) {
    for (int idx = threadIdx.x; idx < (DIM / 2) * DIM; idx += 256) {
        int j   = idx >> 7;          // k-pair index
        int col = idx & (DIM - 1);
        v2f w;
        w.x = W[(2 * j)     * DIM + col];
        w.y = W[(2 * j + 1) * DIM + col];
        shWp[idx] = w;
    }
}

// ---------------------------------------------------------------------------
// Fused: h0 = xin + s/max(cnt,1);  y = h0 @ W + b;  accumulate per-column
// sum / sum-of-squares for BatchNorm. One wave per 16-row tile, full fp32
// precision via V_WMMA_F32_16X16X4_F32; W staged K-pair-packed in LDS (64KB).
//
// Register-blocked: 2 passes x 4 live accumulators. Each K-step is a fenced
// scheduling window of {4 ds_load_b64 -> 4 independent WMMAs}; B loads use
// immediate DS offsets (kk*2048) off one per-lane base.
//
// A-frag layout (16x4 f32, ISA 7.12.2): lanes 0-15 row M=lane hold K=k0,k0+1;
// lanes 16-31 row M=lane-16 hold K=k0+2,k0+3.
// C/D layout: lanes 0-15 col N=lane rows 0-7 in v[0..7]; lanes 16-31 col
// N=lane-16 rows 8-15.
// ---------------------------------------------------------------------------
__global__ __launch_bounds__(256)
void gin_gemm1_kernel(const float* __restrict__ xin,
                      const float* __restrict__ s,
                      const float* __restrict__ cnt,
                      const float* __restrict__ W,
                      const float* __restrict__ bias,
                      float* __restrict__ y,
                      float* __restrict__ gsum,
                      float* __restrict__ gsq) {
    __shared__ v2f shWp[(DIM / 2) * DIM];   // 64 KB
    stage_w_packed(shWp, W);
    __syncthreads();

    const int wave = threadIdx.x >> 5;
    const int lane = threadIdx.x & 31;
    const int half = lane >> 4;      // 0: K pair (0,1), 1: K pair (2,3)
    const int row  = lane & 15;
    const int tile = blockIdx.x * 8 + wave;
    if (tile >= N_TILES) return;     // wave-uniform -> EXEC all-1s for WMMA

    const int node = tile * 16 + row;
    const float inv = 1.0f / fmaxf(cnt[node], 1.0f);

    // Build A fragments in registers: 32 K-steps x v2f (64 floats/lane)
    v2f a[32];
    const float2* xr = (const float2*)(xin + (long long)node * DIM);
    const float2* sr = (const float2*)(s   + (long long)node * DIM);
#pragma unroll
    for (int kk = 0; kk < 32; ++kk) {
        float2 xv = xr[kk * 2 + half];
        float2 sv = sr[kk * 2 + half];
        a[kk].x = xv.x + sv.x * inv;   // (1+eps)*x + mean-agg, eps = 0
        a[kk].y = xv.y + sv.y * inv;
    }

    for (int p = 0; p < 2; ++p) {
        v8f acc[4];
#pragma unroll
        for (int n = 0; n < 4; ++n)
            acc[n] = (v8f){0.f, 0.f, 0.f, 0.f, 0.f, 0.f, 0.f, 0.f};
#pragma unroll
        for (int kk = 0; kk < 32; ++kk) {
#pragma unroll
            for (int n = 0; n < 4; ++n) {
                const int col = p * 64 + n * 16 + row;
                // j = 2*kk + half -> address = (half*DIM + col)*8 + kk*2048
                v2f b = shWp[(2 * kk + half) * DIM + col];
                acc[n] = __builtin_amdgcn_wmma_f32_16x16x4_f32(
                    false, a[kk], false, b, (short)0, acc[n], false, false);
            }
            SCHED_FENCE();
        }
#pragma unroll
        for (int n = 0; n < 4; ++n) {
            const int col = p * 64 + n * 16 + row;
            const float bv = bias[col];
            float lsum = 0.f, lsq = 0.f;
#pragma unroll
            for (int r = 0; r < 8; ++r) {
                float v = acc[n][r] + bv;
                y[(long long)(tile * 16 + half * 8 + r) * DIM + col] = v;
                lsum += v;
                lsq  += v * v;
            }
            atomicAdd(&gsum[col], lsum);
            atomicAdd(&gsq[col],  lsq);
        }
    }
}

// ---------------------------------------------------------------------------
// Fold BN training stats into per-column scale/shift.
// ---------------------------------------------------------------------------
__global__ void gin_bn_kernel(const float* __restrict__ gsum,
                              const float* __restrict__ gsq,
                              const float* __restrict__ gamma,
                              const float* __restrict__ beta,
                              float* __restrict__ scale,
                              float* __restrict__ shift) {
    int i = threadIdx.x;
    if (i < DIM) {
        float mu  = gsum[i] * (1.0f / (float)N_NODES);
        float var = gsq[i]  * (1.0f / (float)N_NODES) - mu * mu;
        var = fmaxf(var, 0.0f);
        float sc = gamma[i] * rsqrtf(var + BN_EPS);
        scale[i] = sc;
        shift[i] = beta[i] - mu * sc;
    }
}

// ---------------------------------------------------------------------------
// Fused: t = relu(y*scale + shift);  out = t @ W + b  (+ optional ReLU
// between layers). Same fenced packed-B WMMA structure as gemm1.
// ---------------------------------------------------------------------------
__global__ __launch_bounds__(256)
void gin_gemm2_kernel(const float* __restrict__ y,
                      const float* __restrict__ scale,
                      const float* __restrict__ shift,
                      const float* __restrict__ W,
                      const float* __restrict__ bias,
                      float* __restrict__ out,
                      int relu_out) {
    __shared__ v2f shWp[(DIM / 2) * DIM];   // 64 KB
    stage_w_packed(shWp, W);
    __syncthreads();

    const int wave = threadIdx.x >> 5;
    const int lane = threadIdx.x & 31;
    const int half = lane >> 4;
    const int row  = lane & 15;
    const int tile = blockIdx.x * 8 + wave;
    if (tile >= N_TILES) return;

    const int node = tile * 16 + row;

    v2f a[32];
    const float2* yr  = (const float2*)(y + (long long)node * DIM);
    const float2* scv = (const float2*)scale;
    const float2* shv = (const float2*)shift;
#pragma unroll
    for (int kk = 0; kk < 32; ++kk) {
        float2 yv = yr[kk * 2 + half];
        float2 sc = scv[kk * 2 + half];
        float2 sh = shv[kk * 2 + half];
        a[kk].x = fmaxf(yv.x * sc.x + sh.x, 0.0f);
        a[kk].y = fmaxf(yv.y * sc.y + sh.y, 0.0f);
    }

    for (int p = 0; p < 2; ++p) {
        v8f acc[4];
#pragma unroll
        for (int n = 0; n < 4; ++n)
            acc[n] = (v8f){0.f, 0.f, 0.f, 0.f, 0.f, 0.f, 0.f, 0.f};
#pragma unroll
        for (int kk = 0; kk < 32; ++kk) {
#pragma unroll
            for (int n = 0; n < 4; ++n) {
                const int col = p * 64 + n * 16 + row;
                v2f b = shWp[(2 * kk + half) * DIM + col];
                acc[n] = __builtin_amdgcn_wmma_f32_16x16x4_f32(
                    false, a[kk], false, b, (short)0, acc[n], false, false);
            }
            SCHED_FENCE();
        }
#pragma unroll
        for (int n = 0; n < 4; ++n) {
            const int col = p * 64 + n * 16 + row;
            const float bv = bias[col];
#pragma unroll
            for (int r = 0; r < 8; ++r) {
                float v = acc[n][r] + bv;
                if (relu_out) v = fmaxf(v, 0.0f);
                out[(long long)(tile * 16 + half * 8 + r) * DIM + col] = v;
            }
        }
    }
}

// ---------------------------------------------------------------------------
// Orchestration. Workspace layout (floats), ordered so the per-layer
// re-zeroed region is one contiguous prefix:
//   s[N*128] | gsum[128] | gsq[128] | cnt[N] | scale[128] | shift[128] | y1[N*128]
// Layer 0 zeroes s..cnt (cnt computed once; degrees are layer-invariant),
// layer 1 zeroes only s..gsq. d_out doubles as the inter-layer feature buffer.
// ---------------------------------------------------------------------------
extern "C" void kernel_launch(void* const* d_in, const int* in_sizes, int n_in,
                              void* d_out, int out_size, void* d_ws, size_t ws_size,
                              hipStream_t stream) {
    const float* x  = (const float*)d_in[0];
    const int*   ei = (const int*)d_in[1];        // edge_index [2, E] (int32 per harness)
    const int* src = ei;
    const int* dst = ei + N_EDGES;
    const float* P[12];
    for (int i = 0; i < 12; ++i) P[i] = (const float*)d_in[2 + i];
    // P: W1_0,b1_0,g_0,be_0,W2_0,b2_0, W1_1,b1_1,g_1,be_1,W2_1,b2_1

    float* ws    = (float*)d_ws;
    float* s     = ws;                                // N*128
    float* gsum  = s + (size_t)N_NODES * DIM;         // 128
    float* gsq   = gsum + DIM;                        // 128
    float* cnt   = gsq + DIM;                         // N
    float* scale = cnt + N_NODES;                     // 128
    float* shift = scale + DIM;                       // 128
    float* y1    = shift + DIM;                       // N*128
    float* out   = (float*)d_out;

    const long long nz0  = (long long)N_NODES * DIM + 2 * DIM + N_NODES; // s..cnt
    const long long nz1  = (long long)N_NODES * DIM + 2 * DIM;           // s..gsq
    const long long n_sc = (long long)N_EDGES * (DIM / 4);
    dim3 sgrid((unsigned)((n_sc + 255) / 256));
    dim3 ggrid((unsigned)((N_TILES + 7) / 8));

    for (int l = 0; l < 2; ++l) {
        const float* in = (l == 0) ? x : out;
        const float* W1 = P[l * 6 + 0];
        const float* b1 = P[l * 6 + 1];
        const float* g  = P[l * 6 + 2];
        const float* be = P[l * 6 + 3];
        const float* W2 = P[l * 6 + 4];
        const float* b2 = P[l * 6 + 5];
        const long long nz = (l == 0) ? nz0 : nz1;

        gin_zero_kernel<<<dim3((unsigned)((nz + 255) / 256)), 256, 0, stream>>>(s, nz);
        gin_scatter_kernel<<<sgrid, 256, 0, stream>>>(in, src, dst, s, cnt,
                                                      (l == 0) ? 1 : 0);
        gin_gemm1_kernel<<<ggrid, 256, 0, stream>>>(in, s, cnt, W1, b1, y1, gsum, gsq);
        gin_bn_kernel<<<dim3(1), 128, 0, stream>>>(gsum, gsq, g, be, scale, shift);
        gin_gemm2_kernel<<<ggrid, 256, 0, stream>>>(y1, scale, shift, W2, b2, out,
                                                    (l == 0) ? 1 : 0);
    }
}